// OnlineClustering_44332652429332
// MI455X (gfx1250) — compile-verified
//
#include <hip/hip_runtime.h>
#include <math.h>

// ---- problem constants (from reference) ----
#define B_CNT   32
#define N_ROWS  784
#define R_TOT   25088          // 32*784
#define K_DIM   1024
#define NK      4096
#define TT_INV  (1.0f/0.06f)   // 1/TARGET_TEMP
#define PT_INV  (1.0f/0.12f)   // 1/PRED_TEMP
#define EXPMAX  50.0f
#define EPS_SK  1e-8f
#define EPS_NRM 1e-7f

typedef float v2f __attribute__((ext_vector_type(2)));
typedef float v8f __attribute__((ext_vector_type(8)));
typedef unsigned int u32x4 __attribute__((ext_vector_type(4)));
typedef int i32x8 __attribute__((ext_vector_type(8)));

// ---------------------------------------------------------------------------
// 1) per-row L2 norm reciprocal: rn[r] = 1 / max(||x_r||, 1e-7)
// ---------------------------------------------------------------------------
__global__ void oc_rownorm(const float* __restrict__ x, float* __restrict__ rn) {
  int r    = (blockIdx.x * blockDim.x + threadIdx.x) >> 5;
  int lane = threadIdx.x & 31;
  if (r >= R_TOT) return;
  const float* row = x + (size_t)r * K_DIM;
  float s = 0.f;
  #pragma unroll 4
  for (int k = lane * 4; k < K_DIM; k += 32 * 4) {
    float4 v = *(const float4*)(row + k);
    s += v.x * v.x + v.y * v.y + v.z * v.z + v.w * v.w;
  }
  #pragma unroll
  for (int off = 16; off; off >>= 1) s += __shfl_xor(s, off, 32);
  if (lane == 0) rn[r] = 1.0f / fmaxf(sqrtf(s), EPS_NRM);
}

// ---------------------------------------------------------------------------
// TDM: 2D tile Global->LDS via tensor_load_to_lds (D# per CDNA5 ISA)
//   tile = tile_d1 rows x tile_d0 elements (4B), row stride = stride0 elems.
//   LDS padding: pad_interval_enc=3 (16 DWORDs), pad_amount_enc=3 (4 DWORDs)
//   -> LDS row stride of 20 floats, matching the WMMA fragment addressing.
// ---------------------------------------------------------------------------
__device__ __forceinline__ void tdm_load_2d_f32(unsigned int lds_off, const float* gptr,
                                                unsigned int tensor_d0, unsigned int tensor_d1,
                                                unsigned int stride0,
                                                unsigned int tile_d0, unsigned int tile_d1,
                                                unsigned int pad_interval_enc,
                                                unsigned int pad_amount_enc) {
  unsigned long long ga = (unsigned long long)(size_t)gptr;
  u32x4 g0;
  g0.x = 1u;                                         // count=1 (user descriptor)
  g0.y = lds_off;                                    // LDS byte address
  g0.z = (unsigned int)ga;                           // global addr [31:0]
  g0.w = (unsigned int)(ga >> 32) | (2u << 30);      // addr[56:32] | type=2 (image)
  i32x8 g1;
  g1[0] = (int)((2u << 16)                           // data_size = 4 bytes
              | (1u << 20)                           // pad_enable
              | (pad_interval_enc << 22)
              | (pad_amount_enc << 25));
  g1[1] = (int)((tensor_d0 & 0xFFFFu) << 16);        // tensor_dim0[15:0] @ bit48
  g1[2] = (int)((tensor_d0 >> 16) | ((tensor_d1 & 0xFFFFu) << 16));
  g1[3] = (int)((tensor_d1 >> 16) | (tile_d0 << 16));// tile_dim0 @ bit112
  g1[4] = (int)tile_d1;                              // tile_dim1 @ bit128 (tile_dim2=0)
  g1[5] = (int)stride0;                              // tensor_dim0_stride[31:0] @ bit160
  g1[6] = 0;                                         // stride0 hi / stride1 lo
  g1[7] = 0;
  asm volatile("tensor_load_to_lds %0, %1" :: "s"(g0), "s"(g1) : "memory");
}

// ---------------------------------------------------------------------------
// 2) f32 WMMA GEMM: logits[r,k] = rn[r] * sum_d x[r,d] * W[k,d]
//    128x128 block tile, 8 waves of 32x64; K staged via TDM, double-buffered.
// ---------------------------------------------------------------------------
#define BM   128
#define BN   128
#define KC   16
#define LDSP 20   // padded row stride (floats): 16B-aligned rows, no bank conflicts
#define NCHUNK (K_DIM / KC)   // 64

__global__ __launch_bounds__(256) void oc_gemm(const float* __restrict__ x,
                                               const float* __restrict__ W,
                                               const float* __restrict__ rn,
                                               float* __restrict__ out) {
  __shared__ __align__(16) float As[2][BM * LDSP];
  __shared__ __align__(16) float Bs[2][BN * LDSP];

  const int tilesN  = NK / BN;                 // 32
  const int rowBase = (blockIdx.x / tilesN) * BM;
  const int colBase = (blockIdx.x % tilesN) * BN;
  const int tid  = threadIdx.x;
  const int wave = tid >> 5, lane = tid & 31;
  const int wm = wave >> 1;                    // 0..3 -> M offset wm*32
  const int wn = wave & 1;                     // 0..1 -> N offset wn*64
  const int lrow    = lane & 15;
  const int halfsel = (lane >> 4) << 1;        // 0 or 2 (K pair select per ISA layout)

  v8f acc[2][4];
  #pragma unroll
  for (int i = 0; i < 2; ++i)
    #pragma unroll
    for (int j = 0; j < 4; ++j)
      acc[i][j] = (v8f){0.f, 0.f, 0.f, 0.f, 0.f, 0.f, 0.f, 0.f};

  // prologue: TDM chunk 0 into buffer 0 (wave 0 owns the tensor engine)
  if (wave == 0) {
    tdm_load_2d_f32((unsigned int)(size_t)&As[0][0], x + (size_t)rowBase * K_DIM,
                    K_DIM, R_TOT, K_DIM, KC, BM, 3u, 3u);
    tdm_load_2d_f32((unsigned int)(size_t)&Bs[0][0], W + (size_t)colBase * K_DIM,
                    K_DIM, NK, K_DIM, KC, BN, 3u, 3u);
  }

  for (int c = 0; c < NCHUNK; ++c) {
    const int buf = c & 1;
    if (wave == 0) {
      if (c + 1 < NCHUNK) {
        const int kc1 = (c + 1) * KC;
        tdm_load_2d_f32((unsigned int)(size_t)&As[buf ^ 1][0],
                        x + (size_t)rowBase * K_DIM + kc1,
                        K_DIM, R_TOT, K_DIM, KC, BM, 3u, 3u);
        tdm_load_2d_f32((unsigned int)(size_t)&Bs[buf ^ 1][0],
                        W + (size_t)colBase * K_DIM + kc1,
                        K_DIM, NK, K_DIM, KC, BN, 3u, 3u);
        __builtin_amdgcn_s_wait_tensorcnt(2);   // chunk c complete (in-order)
      } else {
        __builtin_amdgcn_s_wait_tensorcnt(0);
      }
    }
    __syncthreads();   // chunk c visible in LDS to all waves

    const float* Ab = As[buf];
    const float* Bb = Bs[buf];
    #pragma unroll
    for (int kk = 0; kk < KC; kk += 4) {
      v2f af[2], bf[4];
      #pragma unroll
      for (int ti = 0; ti < 2; ++ti)
        af[ti] = *(const v2f*)(&Ab[(wm * 32 + ti * 16 + lrow) * LDSP + kk + halfsel]);
      #pragma unroll
      for (int tj = 0; tj < 4; ++tj)
        bf[tj] = *(const v2f*)(&Bb[(wn * 64 + tj * 16 + lrow) * LDSP + kk + halfsel]);
      #pragma unroll
      for (int ti = 0; ti < 2; ++ti)
        #pragma unroll
        for (int tj = 0; tj < 4; ++tj)
          acc[ti][tj] = __builtin_amdgcn_wmma_f32_16x16x4_f32(
              false, af[ti], false, bf[tj], (short)0, acc[ti][tj], false, false);
    }
    __syncthreads();   // compute done before buffer is overwritten
  }

  // epilogue: scale C rows by rn and store
  const int rowHalf = (lane >> 4) * 8;
  #pragma unroll
  for (int ti = 0; ti < 2; ++ti)
    #pragma unroll
    for (int tj = 0; tj < 4; ++tj) {
      int col = colBase + wn * 64 + tj * 16 + lrow;
      #pragma unroll
      for (int i = 0; i < 8; ++i) {
        int row = rowBase + wm * 32 + ti * 16 + i + rowHalf;
        out[(size_t)row * NK + col] = acc[ti][tj][i] * rn[row];
      }
    }
}

// ---------------------------------------------------------------------------
// 3) shift[b,k] = max_n logits[b,n,k] / TARGET_TEMP
// ---------------------------------------------------------------------------
__global__ void oc_colmax(const float* __restrict__ logits, float* __restrict__ shift) {
  int t = blockIdx.x * blockDim.x + threadIdx.x;   // 131072 threads
  int b = t >> 12, k = t & (NK - 1);
  const float* base = logits + (size_t)b * N_ROWS * NK + k;
  float m = -3.4e38f;
  for (int n = 0; n < N_ROWS; ++n) m = fmaxf(m, base[(size_t)n * NK]);
  shift[t] = m * TT_INV;
}

__global__ void oc_init_uv(float* __restrict__ u, float* __restrict__ v) {
  int t = blockIdx.x * blockDim.x + threadIdx.x;
  if (t < R_TOT) u[t] = 1.f;
  if (t < B_CNT * NK) v[t] = 1.f;
}

// ---------------------------------------------------------------------------
// 4) column step: S_k = sum_n u_n * M0[n,k];  v_k <- v_k / (v_k*S_k + eps)
// ---------------------------------------------------------------------------
__global__ void oc_colstep(const float* __restrict__ logits, const float* __restrict__ shift,
                           const float* __restrict__ u, float* __restrict__ v) {
  int t = blockIdx.x * blockDim.x + threadIdx.x;
  int b = t >> 12;
  const float* base = logits + (size_t)b * N_ROWS * NK + (t & (NK - 1));
  const float* ub = u + b * N_ROWS;
  float sh = EXPMAX - shift[t];
  float S = 0.f;
  for (int n = 0; n < N_ROWS; ++n)
    S += ub[n] * __expf(base[(size_t)n * NK] * TT_INV + sh);
  float vv = v[t];
  v[t] = vv / (vv * S + EPS_SK);
}

// ---------------------------------------------------------------------------
// 5) row step: T_n = sum_k M0[n,k]*v_k;  u_n <- u_n / (u_n*T_n + eps)
// ---------------------------------------------------------------------------
__global__ void oc_rowstep(const float* __restrict__ logits, const float* __restrict__ shift,
                           float* __restrict__ u, const float* __restrict__ v) {
  int r    = (blockIdx.x * blockDim.x + threadIdx.x) >> 5;
  int lane = threadIdx.x & 31;
  if (r >= R_TOT) return;
  int b = r / N_ROWS;
  const float* row = logits + (size_t)r * NK;
  const float* shb = shift + b * NK;
  const float* vb  = v + b * NK;
  float T = 0.f;
  for (int k = lane; k < NK; k += 32)
    T += __expf(row[k] * TT_INV + EXPMAX - shb[k]) * vb[k];
  #pragma unroll
  for (int off = 16; off; off >>= 1) T += __shfl_xor(T, off, 32);
  if (lane == 0) { float uu = u[r]; u[r] = uu / (uu * T + EPS_SK); }
}

// ---------------------------------------------------------------------------
// 6) final: A = u*M0*v written in-place over logits; fused per-row CE loss
// ---------------------------------------------------------------------------
__global__ void oc_final(float* __restrict__ logits, const float* __restrict__ shift,
                         const float* __restrict__ u, const float* __restrict__ v,
                         float* __restrict__ rowloss) {
  int r    = (blockIdx.x * blockDim.x + threadIdx.x) >> 5;
  int lane = threadIdx.x & 31;
  if (r >= R_TOT) return;
  int b = r / N_ROWS;
  float* row = logits + (size_t)r * NK;
  const float* shb = shift + b * NK;
  const float* vb  = v + b * NK;
  float uu = u[r];

  // online logsumexp of logits/PRED_TEMP
  float m = -3.4e38f, s = 0.f;
  for (int k = lane; k < NK; k += 32) {
    float p = row[k] * PT_INV;
    float nm = fmaxf(m, p);
    s = s * __expf(m - nm) + __expf(p - nm);
    m = nm;
  }
  #pragma unroll
  for (int off = 16; off; off >>= 1) {
    float om = __shfl_xor(m, off, 32);
    float os = __shfl_xor(s, off, 32);
    float nm = fmaxf(m, om);
    s = s * __expf(m - nm) + os * __expf(om - nm);
    m = nm;
  }
  float lse = m + __logf(s);

  float sA = 0.f, sAl = 0.f;
  for (int k = lane; k < NK; k += 32) {
    float l = row[k];
    float A = uu * __expf(l * TT_INV + EXPMAX - shb[k]) * vb[k];
    row[k] = A;                      // in-place: logits -> assignments
    sA += A; sAl += A * l;
  }
  #pragma unroll
  for (int off = 16; off; off >>= 1) {
    sA  += __shfl_xor(sA, off, 32);
    sAl += __shfl_xor(sAl, off, 32);
  }
  if (lane == 0) rowloss[r] = -(sAl * PT_INV - lse * sA);
}

__global__ void oc_loss_reduce(const float* __restrict__ rowloss, float* __restrict__ lossOut) {
  __shared__ float sm[256];
  float s = 0.f;
  for (int i = threadIdx.x; i < R_TOT; i += 256) s += rowloss[i];
  sm[threadIdx.x] = s;
  __syncthreads();
  for (int off = 128; off; off >>= 1) {
    if (threadIdx.x < off) sm[threadIdx.x] += sm[threadIdx.x + off];
    __syncthreads();
  }
  if (threadIdx.x == 0) lossOut[0] = sm[0] / (float)R_TOT;
}

// ---------------------------------------------------------------------------
extern "C" void kernel_launch(void* const* d_in, const int* in_sizes, int n_in,
                              void* d_out, int out_size, void* d_ws, size_t ws_size,
                              hipStream_t stream) {
  const float* x = (const float*)d_in[0];
  const float* W = (const float*)d_in[1];
  float* out = (float*)d_out;                // [25088*4096] assignments, then [1] loss

  float* ws      = (float*)d_ws;             // ~1.35 MB total scratch
  float* rn      = ws;                       // 25088
  float* shift   = rn + R_TOT;               // 131072
  float* u       = shift + B_CNT * NK;       // 25088
  float* v       = u + R_TOT;                // 131072
  float* rowloss = v + B_CNT * NK;           // 25088

  oc_rownorm<<<R_TOT * 32 / 256, 256, 0, stream>>>(x, rn);
  oc_gemm<<<(R_TOT / BM) * (NK / BN), 256, 0, stream>>>(x, W, rn, out);
  oc_colmax<<<B_CNT * NK / 256, 256, 0, stream>>>(out, shift);
  oc_init_uv<<<B_CNT * NK / 256, 256, 0, stream>>>(u, v);
  for (int it = 0; it < 3; ++it) {
    oc_colstep<<<B_CNT * NK / 256, 256, 0, stream>>>(out, shift, u, v);
    oc_rowstep<<<R_TOT * 32 / 256, 256, 0, stream>>>(out, shift, u, v);
  }
  oc_final<<<R_TOT * 32 / 256, 256, 0, stream>>>(out, shift, u, v, rowloss);
  oc_loss_reduce<<<1, 256, 0, stream>>>(rowloss, out + (size_t)R_TOT * NK);
}